// Sliding_Full_connection_1640677507426
// MI455X (gfx1250) — compile-verified
//
#include <hip/hip_runtime.h>

typedef __attribute__((ext_vector_type(2))) float v2f;
typedef __attribute__((ext_vector_type(8))) float v8f;

#define H 512
#define W 512
#define CIN 3
#define COUT 2
#define LDS_NROWS 12          // 3 channels x 4 input rows (h0-1 .. h0+2)
#define LDS_ROWW  520         // padded LDS row pitch (floats)
// layout: input col icol -> lds col icol+4 (halo cols 3 and 516 are zero)

__global__ __launch_bounds__(256) void conv3x3_wmma_f32(
    const float* __restrict__ x,      // (64,3,512,512)
    const float* __restrict__ wgt,    // (2,27) k = c*9 + dh*3 + dw
    const float* __restrict__ bias,   // (2,)
    float* __restrict__ out)          // (64,2,512,512)
{
    const int h0  = blockIdx.x * 2;   // first of two output rows
    const int b   = blockIdx.y;      // batch
    const int tid = threadIdx.x;

    __shared__ float lds[LDS_NROWS * LDS_ROWW];

    // ---- Zero the two halo columns of each staged row ----
    if (tid < 2 * LDS_NROWS) {
        int r = tid >> 1;
        lds[r * LDS_ROWW + ((tid & 1) ? 516 : 3)] = 0.0f;
    }

    // ---- Stage 3 channels x 4 rows (h0-1 .. h0+2), vectorized float4 ----
    // 12 rows x 128 float4 each; out-of-range rows staged as zeros.
    for (int idx = tid; idx < LDS_NROWS * 128; idx += 256) {
        int r   = idx >> 7;           // 0..11 : r = c*4 + dh4
        int q   = idx & 127;          // float4 index within row
        int c   = r >> 2;
        int dh4 = r & 3;
        int row = h0 + dh4 - 1;
        float4 v = make_float4(0.0f, 0.0f, 0.0f, 0.0f);
        if (row >= 0 && row < H)
            v = *(const float4*)(x + (((size_t)b * CIN + c) * H + row) * W + q * 4);
        *(float4*)(&lds[r * LDS_ROWW + 4 + q * 4]) = v;   // 16B-aligned
    }
    __syncthreads();

    const int lane = tid & 31;
    const int wave = tid >> 5;
    const int m    = lane & 15;       // M index (pixel within segment) / N index
    const int half = lane >> 4;       // K sub-column selector

    // ---- B fragments: B[k][n] = wgt[n][k] for n<2, zero otherwise (k=27 pad=0) ----
    v2f bfrag[7];
#pragma unroll
    for (int j = 0; j < 7; ++j) {
        float b0 = 0.0f, b1 = 0.0f;
        int k0 = j * 4 + 2 * half;
        if (m < COUT) {
            b0 = wgt[m * 27 + k0];                  // k0 <= 26 always
            if (k0 + 1 < 27) b1 = wgt[m * 27 + k0 + 1];
        }
        bfrag[j][0] = b0;
        bfrag[j][1] = b1;
    }

    // ---- Per-lane LDS tap offsets (row 0); row 1 adds LDS_ROWW ----
    int offA[14];
#pragma unroll
    for (int j = 0; j < 7; ++j) {
#pragma unroll
        for (int v = 0; v < 2; ++v) {
            int k = j * 4 + 2 * half + v;
            if (k > 26) k = 26;                     // B is zero there; value unused
            int c  = k / 9;
            int r  = k - c * 9;
            int dh = r / 3;
            int dw = r - dh * 3;
            // input col (w0+m) + dw - 1  ->  lds col (w0+m) + dw + 3
            offA[j * 2 + v] = (c * 4 + dh) * LDS_ROWW + m + dw + 3;
        }
    }

    const float bv0 = bias[0];
    const float bv1 = bias[1];

    // ---- Two output rows; each row tiled by 8 waves x 4 segments of 16 ----
#pragma unroll
    for (int rr = 0; rr < 2; ++rr) {
        const float* ldsr = lds + rr * LDS_ROWW;    // shift all taps down one row
        const int h = h0 + rr;

#pragma unroll
        for (int s = 0; s < 4; ++s) {
            const int w0 = (wave * 4 + s) * 16;

            v8f acc = {};
#pragma unroll
            for (int j = 0; j < 7; ++j) {
                v2f a;
                a[0] = ldsr[offA[j * 2 + 0] + w0];
                a[1] = ldsr[offA[j * 2 + 1] + w0];
                acc = __builtin_amdgcn_wmma_f32_16x16x4_f32(
                    /*neg_a=*/false, a, /*neg_b=*/false, bfrag[j],
                    /*c_mod=*/(short)0, acc, /*reuse_a=*/false, /*reuse_b=*/false);
            }

            // D[M][N]: lanes {0,1,16,17} hold N(channel) 0/1, M = r + 8*(lane>>4)
            if ((lane & 14) == 0) {
                const int ch = lane & 1;
                const int mb = (lane >> 4) * 8;
                const float bv = ch ? bv1 : bv0;
                float* o = out + ((((size_t)b * COUT + ch) * H + h) * W + w0 + mb);
                float4 lo = make_float4(acc[0] + bv, acc[1] + bv, acc[2] + bv, acc[3] + bv);
                float4 hi = make_float4(acc[4] + bv, acc[5] + bv, acc[6] + bv, acc[7] + bv);
                *(float4*)(o + 0) = lo;
                *(float4*)(o + 4) = hi;
            }
        }
    }
}

extern "C" void kernel_launch(void* const* d_in, const int* in_sizes, int n_in,
                              void* d_out, int out_size, void* d_ws, size_t ws_size,
                              hipStream_t stream) {
    const float* x    = (const float*)d_in[0];
    const float* wgt  = (const float*)d_in[1];
    const float* bias = (const float*)d_in[2];
    float* out        = (float*)d_out;

    dim3 grid(H / 2, 64, 1);   // (pair of output rows, batch)
    dim3 block(256, 1, 1);     // 8 wave32
    conv3x3_wmma_f32<<<grid, block, 0, stream>>>(x, wgt, bias, out);
}